// OCSolver_58428735095166
// MI455X (gfx1250) — compile-verified
//
#include <hip/hip_runtime.h>
#include <hip/hip_bf16.h>
#include <cstdint>

#define T_STEPS 25
#define NITER   20
#define DTC     0.1f
#define REGC    1e-6f

#define LANES   32
// LDS slots per lane: X (T+1)*3 = 78, U 75, k 75, K 225  -> 453 dwords/lane
#define SLOT_X  0
#define SLOT_U  78
#define SLOT_K  153
#define SLOT_KK 228
#define SLOTS   453
#define HSTAGE  1344            // H staging (32*36 dwords), after input staging [0,1344)

#if __has_builtin(__builtin_amdgcn_global_load_async_to_lds_b32) && \
    __has_builtin(__builtin_amdgcn_global_store_async_from_lds_b32) && \
    __has_builtin(__builtin_amdgcn_s_wait_asynccnt)
#define USE_ASYNC 1
#else
#define USE_ASYNC 0
#endif

#if __has_builtin(__builtin_amdgcn_wmma_f32_16x16x4_f32)
#define USE_WMMA 1
#else
#define USE_WMMA 0
#endif

typedef float v2f __attribute__((ext_vector_type(2)));
typedef float v8f __attribute__((ext_vector_type(8)));

// Address-space-qualified pointer casts for the async-LDS builtins (params are
// int ptrs in AS(1)/AS(3) per hipcc's diagnostic).
#define AS1INT(p) ((__attribute__((address_space(1))) int*)(__attribute__((address_space(1))) void*)(uintptr_t)(p))
#define AS3INT(p) ((__attribute__((address_space(3))) int*)(__attribute__((address_space(3))) void*)((void*)(p)))

#define LDSV(slot) lds[(slot)*LANES + lane]

__global__ __launch_bounds__(LANES) void ilqr_cdna5_kernel(
    const float* __restrict__ in, float* __restrict__ out, int n)
{
    __shared__ float lds[SLOTS * LANES];
    const int lane = threadIdx.x;
    const int blk  = blockIdx.x;

    // ---------------- stage inputs, coalesced, via async-to-LDS -------------
    const float* gin = in + (size_t)blk * (LANES * 42);
    const int valid_in = n * 42 - blk * (LANES * 42);   // elements this block may read
#pragma unroll 1
    for (int i = 0; i < 42; ++i) {
        int j  = i * LANES + lane;
        int jj = (j < valid_in) ? j : 0;                // clamp, keep lanes uniform
#if USE_ASYNC
        __builtin_amdgcn_global_load_async_to_lds_b32(AS1INT(gin + jj), AS3INT(&lds[j]), 0, 0);
#else
        lds[j] = gin[jj];
#endif
    }
#if USE_ASYNC
    __builtin_amdgcn_s_wait_asynccnt(0);
#endif
    __syncthreads();

    // ---------------- H = 2*P^T P (exact f32) --------------------------------
    float Hm[36], bv[6];
#if USE_WMMA
    {
        // Two problems per 16x16x4 f32 WMMA (diagonal 6x6 blocks of C are the
        // per-problem P^T P; lanes 16-21 read the off-diagonal block C[8..13][0..5]).
        // A[m][k] = P_p[k][m'], B[k][n] = P_p[k][n'] -> identical per-lane
        // fragment values, so one register pair serves as both A and B.
        const int half = lane >> 4;          // A/B K-half select
        const int l15  = lane & 15;          // M (A) == N (B) index
        const bool vm  = (l15 < 6) | ((l15 >= 8) & (l15 < 14));
        const int psel = (l15 >= 8) ? 1 : 0; // which problem of the pair
        const int ij   = l15 & 7;            // row/col inside the 6x6 block
#pragma unroll 1
        for (int pr = 0; pr < 16; ++pr) {
            const int p0   = pr * 2;
            const int base = (p0 + psel) * 42 + ij;       // + k*6 indexes P_p[k][ij]
            v2f a0, a1;
#pragma unroll
            for (int v = 0; v < 2; ++v) {
                int k0 = half * 2 + v;                    // 0..3 (always < 6)
                int k1 = 4 + half * 2 + v;                // 4..7 (pad k>=6 with 0)
                a0[v] = vm ? lds[base + k0 * 6] : 0.0f;
                a1[v] = (vm && k1 < 6) ? lds[base + k1 * 6] : 0.0f;
            }
            v8f c = {};
            c = __builtin_amdgcn_wmma_f32_16x16x4_f32(false, a0, false, a0,
                                                      (short)0, c, false, false);
            c = __builtin_amdgcn_wmma_f32_16x16x4_f32(false, a1, false, a1,
                                                      (short)0, c, false, false);
            // C layout: lane(N)=l15, VGPR r -> M=r (lanes 0-15) / M=8+r (16-31)
            // => lanes 0-5: p0's H[r][l15]; lanes 16-21: C[8+r][l15] = (P_{p0+1}^T P_{p0})[r][l15].
            if (l15 < 6) {
                const int pdst = p0 + half;
#pragma unroll
                for (int r = 0; r < 6; ++r)
                    lds[HSTAGE + pdst * 36 + r * 6 + l15] = c[r];
            }
        }
        __syncthreads();
#pragma unroll
        for (int i = 0; i < 36; ++i) Hm[i] = 2.0f * lds[HSTAGE + lane * 36 + i];
    }
#else
    {
        float PP[36];
        const int base = lane * 42;
#pragma unroll
        for (int i = 0; i < 36; ++i) PP[i] = lds[base + i];
#pragma unroll
        for (int i = 0; i < 6; ++i)
#pragma unroll
            for (int j = 0; j < 6; ++j) {
                float acc = 0.f;
#pragma unroll
                for (int k = 0; k < 6; ++k) acc += PP[k*6+i] * PP[k*6+j];
                Hm[i*6+j] = 2.0f * acc;
            }
    }
#endif
    Hm[21] += 2.0f; Hm[28] += 2.0f; Hm[35] += 2.0f;   // H[3:,3:] += 2*I
#pragma unroll
    for (int i = 0; i < 6; ++i) bv[i] = lds[lane * 42 + 36 + i];
    bv[3] -= 2.0f;                                     // -2*W_REF*U_DES

    __syncthreads();
    // zero X (slots 0..77) and U (78..152): initial rollout with K=k=0 gives all-zeros
#pragma unroll 1
    for (int i = 0; i < 153; ++i) lds[i*LANES + lane] = 0.0f;
    __syncthreads();

    // ---------------- iLQR iterations ---------------------------------------
    for (int it = 0; it < NITER; ++it) {
        // terminal cost-to-go
        float Vx[3], Vxx[9];
        {
            float xt0 = LDSV(SLOT_X + T_STEPS*3 + 0);
            float xt1 = LDSV(SLOT_X + T_STEPS*3 + 1);
            float xt2 = LDSV(SLOT_X + T_STEPS*3 + 2);
#pragma unroll
            for (int i = 0; i < 3; ++i)
                Vx[i] = Hm[i*6+0]*xt0 + Hm[i*6+1]*xt1 + Hm[i*6+2]*xt2 + bv[i];
#pragma unroll
            for (int i = 0; i < 3; ++i)
#pragma unroll
                for (int j = 0; j < 3; ++j)
                    Vxx[i*3+j] = Hm[i*6+j];
        }

        // -------- backward Riccati sweep (fx/fu/cx/cu recomputed on the fly)
#pragma unroll 1
        for (int t = T_STEPS - 1; t >= 0; --t) {
            float x0 = LDSV(SLOT_X + t*3 + 0);
            float x1 = LDSV(SLOT_X + t*3 + 1);
            float x2 = LDSV(SLOT_X + t*3 + 2);
            float u0 = LDSV(SLOT_U + t*3 + 0);
            float u1 = LDSV(SLOT_U + t*3 + 1);
            float u2 = LDSV(SLOT_U + t*3 + 2);
            float s, c; __sincosf(x2, &s, &c);
            float a02 = DTC * (-s*u0 - c*u1);          // fx[0][2]
            float a12 = DTC * ( c*u0 - s*u1);          // fx[1][2]

            float g[6];
#pragma unroll
            for (int j = 0; j < 6; ++j)
                g[j] = x0*Hm[0*6+j] + x1*Hm[1*6+j] + x2*Hm[2*6+j]
                     + u0*Hm[3*6+j] + u1*Hm[4*6+j] + u2*Hm[5*6+j] + bv[j];

            float Qx0 = g[0] + Vx[0];
            float Qx1 = g[1] + Vx[1];
            float Qx2 = g[2] + a02*Vx[0] + a12*Vx[1] + Vx[2];
            float Qu[3];
            Qu[0] = g[3] + DTC*( c*Vx[0] + s*Vx[1]);
            Qu[1] = g[4] + DTC*(-s*Vx[0] + c*Vx[1]);
            Qu[2] = g[5] + DTC*Vx[2];

            float W[9], Z[9];                          // W = Vxx@fx, Z = Vxx@fu
#pragma unroll
            for (int i = 0; i < 3; ++i) {
                W[i*3+0] = Vxx[i*3+0];
                W[i*3+1] = Vxx[i*3+1];
                W[i*3+2] = Vxx[i*3+0]*a02 + Vxx[i*3+1]*a12 + Vxx[i*3+2];
                Z[i*3+0] = DTC*( Vxx[i*3+0]*c + Vxx[i*3+1]*s);
                Z[i*3+1] = DTC*(-Vxx[i*3+0]*s + Vxx[i*3+1]*c);
                Z[i*3+2] = DTC*  Vxx[i*3+2];
            }
            float Qxx[9], Quu[9], Qux[9];
#pragma unroll
            for (int j = 0; j < 3; ++j) {
                Qxx[0*3+j] = Hm[0*6+j] + W[0*3+j];
                Qxx[1*3+j] = Hm[1*6+j] + W[1*3+j];
                Qxx[2*3+j] = Hm[2*6+j] + a02*W[0*3+j] + a12*W[1*3+j] + W[2*3+j];
                Quu[0*3+j] = Hm[3*6+3+j] + DTC*( c*Z[0*3+j] + s*Z[1*3+j]);
                Quu[1*3+j] = Hm[4*6+3+j] + DTC*(-s*Z[0*3+j] + c*Z[1*3+j]);
                Quu[2*3+j] = Hm[5*6+3+j] + DTC*Z[2*3+j];
                Qux[0*3+j] = Hm[3*6+j]   + DTC*( c*W[0*3+j] + s*W[1*3+j]);
                Qux[1*3+j] = Hm[4*6+j]   + DTC*(-s*W[0*3+j] + c*W[1*3+j]);
                Qux[2*3+j] = Hm[5*6+j]   + DTC*W[2*3+j];
            }
            Quu[0] += REGC; Quu[4] += REGC; Quu[8] += REGC;

            // 3x3 inverse via adjugate (replaces linalg.solve)
            float inv[9];
            inv[0] = Quu[4]*Quu[8] - Quu[5]*Quu[7];
            inv[1] = Quu[2]*Quu[7] - Quu[1]*Quu[8];
            inv[2] = Quu[1]*Quu[5] - Quu[2]*Quu[4];
            inv[3] = Quu[5]*Quu[6] - Quu[3]*Quu[8];
            inv[4] = Quu[0]*Quu[8] - Quu[2]*Quu[6];
            inv[5] = Quu[2]*Quu[3] - Quu[0]*Quu[5];
            inv[6] = Quu[3]*Quu[7] - Quu[4]*Quu[6];
            inv[7] = Quu[1]*Quu[6] - Quu[0]*Quu[7];
            inv[8] = Quu[0]*Quu[4] - Quu[1]*Quu[3];
            float det  = Quu[0]*inv[0] + Quu[1]*inv[3] + Quu[2]*inv[6];
            float rdet = 1.0f / det;
#pragma unroll
            for (int i = 0; i < 9; ++i) inv[i] *= rdet;

            float kt[3], Kt[9];
#pragma unroll
            for (int i = 0; i < 3; ++i)
                kt[i] = -(inv[i*3+0]*Qu[0] + inv[i*3+1]*Qu[1] + inv[i*3+2]*Qu[2]);
#pragma unroll
            for (int i = 0; i < 3; ++i)
#pragma unroll
                for (int j = 0; j < 3; ++j)
                    Kt[i*3+j] = -(inv[i*3+0]*Qux[0*3+j] + inv[i*3+1]*Qux[1*3+j]
                                + inv[i*3+2]*Qux[2*3+j]);

            // Vx_n = Qx + Kt^T(Quu kt + Qu) + Qux^T kt
            float tv[3];
#pragma unroll
            for (int j = 0; j < 3; ++j)
                tv[j] = Quu[j*3+0]*kt[0] + Quu[j*3+1]*kt[1] + Quu[j*3+2]*kt[2] + Qu[j];
            float Vxn[3] = {Qx0, Qx1, Qx2};
#pragma unroll
            for (int i = 0; i < 3; ++i) {
                float acc = 0.f;
#pragma unroll
                for (int j = 0; j < 3; ++j)
                    acc += Kt[j*3+i]*tv[j] + Qux[j*3+i]*kt[j];
                Vxn[i] += acc;
            }
            // Vxx_n = Qxx + Kt^T Quu Kt + Kt^T Qux + Qux^T Kt ; symmetrize
            float Y[9];
#pragma unroll
            for (int j = 0; j < 3; ++j)
#pragma unroll
                for (int m = 0; m < 3; ++m)
                    Y[j*3+m] = Quu[j*3+0]*Kt[0*3+m] + Quu[j*3+1]*Kt[1*3+m]
                             + Quu[j*3+2]*Kt[2*3+m];
            float Vn[9];
#pragma unroll
            for (int i = 0; i < 3; ++i)
#pragma unroll
                for (int j = 0; j < 3; ++j) {
                    float acc = Qxx[i*3+j];
#pragma unroll
                    for (int l = 0; l < 3; ++l)
                        acc += Kt[l*3+i]*(Y[l*3+j] + Qux[l*3+j]) + Qux[l*3+i]*Kt[l*3+j];
                    Vn[i*3+j] = acc;
                }
#pragma unroll
            for (int i = 0; i < 3; ++i)
#pragma unroll
                for (int j = 0; j < 3; ++j)
                    Vxx[i*3+j] = 0.5f*(Vn[i*3+j] + Vn[j*3+i]);
            Vx[0] = Vxn[0]; Vx[1] = Vxn[1]; Vx[2] = Vxn[2];

#pragma unroll
            for (int i = 0; i < 3; ++i) LDSV(SLOT_K  + t*3 + i) = kt[i];
#pragma unroll
            for (int i = 0; i < 9; ++i) LDSV(SLOT_KK + t*9 + i) = Kt[i];
        }

        // -------- forward rollout with alpha = 1 (in-place update of X, U)
        float xs0 = 0.f, xs1 = 0.f, xs2 = 0.f;
#pragma unroll 1
        for (int t = 0; t < T_STEPS; ++t) {
            float xr0 = LDSV(SLOT_X + t*3 + 0);
            float xr1 = LDSV(SLOT_X + t*3 + 1);
            float xr2 = LDSV(SLOT_X + t*3 + 2);
            float ur0 = LDSV(SLOT_U + t*3 + 0);
            float ur1 = LDSV(SLOT_U + t*3 + 1);
            float ur2 = LDSV(SLOT_U + t*3 + 2);
            float d0 = xs0 - xr0, d1 = xs1 - xr1, d2 = xs2 - xr2;
            float uu0 = ur0 + LDSV(SLOT_K + t*3 + 0)
                      + LDSV(SLOT_KK+t*9+0)*d0 + LDSV(SLOT_KK+t*9+1)*d1 + LDSV(SLOT_KK+t*9+2)*d2;
            float uu1 = ur1 + LDSV(SLOT_K + t*3 + 1)
                      + LDSV(SLOT_KK+t*9+3)*d0 + LDSV(SLOT_KK+t*9+4)*d1 + LDSV(SLOT_KK+t*9+5)*d2;
            float uu2 = ur2 + LDSV(SLOT_K + t*3 + 2)
                      + LDSV(SLOT_KK+t*9+6)*d0 + LDSV(SLOT_KK+t*9+7)*d1 + LDSV(SLOT_KK+t*9+8)*d2;
            LDSV(SLOT_X + t*3+0) = xs0; LDSV(SLOT_X + t*3+1) = xs1; LDSV(SLOT_X + t*3+2) = xs2;
            LDSV(SLOT_U + t*3+0) = uu0; LDSV(SLOT_U + t*3+1) = uu1; LDSV(SLOT_U + t*3+2) = uu2;
            float s, c; __sincosf(xs2, &s, &c);
            xs0 += DTC*(c*uu0 - s*uu1);
            xs1 += DTC*(s*uu0 + c*uu1);
            xs2 += DTC*uu2;
        }
        LDSV(SLOT_X + T_STEPS*3+0) = xs0;
        LDSV(SLOT_X + T_STEPS*3+1) = xs1;
        LDSV(SLOT_X + T_STEPS*3+2) = xs2;
    }

    __syncthreads();
    // ---------------- coalesced output via LDS-transposed async stores ------
    float* gout = out + (size_t)blk * (LANES * 75);
    const int out_lim = n * 75 - blk * (LANES * 75);
#pragma unroll 1
    for (int i = 0; i < 75; ++i) {
        int j = i * LANES + lane;
        if (j < out_lim) {
            int pp = j / 75;
            int cc = j - pp * 75;
#if USE_ASYNC
            __builtin_amdgcn_global_store_async_from_lds_b32(
                AS1INT(gout + j), AS3INT(&lds[(SLOT_U + cc)*LANES + pp]), 0, 0);
#else
            gout[j] = lds[(SLOT_U + cc)*LANES + pp];
#endif
        }
    }
#if USE_ASYNC
    __builtin_amdgcn_s_wait_asynccnt(0);
#endif
}

extern "C" void kernel_launch(void* const* d_in, const int* in_sizes, int n_in,
                              void* d_out, int out_size, void* d_ws, size_t ws_size,
                              hipStream_t stream) {
    const float* in = (const float*)d_in[0];
    float* out = (float*)d_out;
    const int n = in_sizes[0] / 42;          // number of problems (4096)
    const int blocks = (n + LANES - 1) / LANES;
    ilqr_cdna5_kernel<<<dim3(blocks), dim3(LANES), 0, stream>>>(in, out, n);
}